// Quant_ReLU6_55671366090844
// MI455X (gfx1250) — compile-verified
//
#include <hip/hip_runtime.h>
#include <hip/hip_bf16.h>
#include <stdint.h>

#ifndef QR6_FLT_MAX
#define QR6_FLT_MAX 3.402823466e+38f
#endif

typedef __attribute__((ext_vector_type(4))) float v4f;

constexpr int kNumSamples      = 64;
constexpr int kNPer            = 256 * 56 * 56;   // 802816 elems / sample
constexpr int kV4Per           = kNPer / 4;       // 200704 float4 / sample
constexpr int kThreads         = 256;             // 8 waves (wave32)
constexpr int kV4PerThread     = 8;
constexpr int kBlocksPerSample = kV4Per / (kThreads * kV4PerThread); // 98 (exact)

// ---------------------------------------------------------------------------
// Pass 1: per-block partial min/max. 128-bit vector loads, wave32 shuffle
// reduction, cross-wave LDS reduction.
// ---------------------------------------------------------------------------
__global__ __launch_bounds__(kThreads) void qr6_partial_minmax(
    const float* __restrict__ x, float* __restrict__ partial) {
  const int s   = (int)blockIdx.x / kBlocksPerSample;
  const int blk = (int)blockIdx.x % kBlocksPerSample;
  const v4f* __restrict__ p = reinterpret_cast<const v4f*>(x) +
      (size_t)s * kV4Per + (size_t)blk * (kThreads * kV4PerThread);

  float mn =  QR6_FLT_MAX;
  float mx = -QR6_FLT_MAX;
#pragma unroll
  for (int i = 0; i < kV4PerThread; ++i) {
    v4f v = p[i * kThreads + threadIdx.x];
    mn = fminf(mn, fminf(fminf(v.x, v.y), fminf(v.z, v.w)));
    mx = fmaxf(mx, fmaxf(fmaxf(v.x, v.y), fmaxf(v.z, v.w)));
  }
  // wave32 butterfly reduction
#pragma unroll
  for (int off = 16; off > 0; off >>= 1) {
    mn = fminf(mn, __shfl_xor(mn, off, 32));
    mx = fmaxf(mx, __shfl_xor(mx, off, 32));
  }
  __shared__ float smn[kThreads / 32];
  __shared__ float smx[kThreads / 32];
  const int wave = threadIdx.x >> 5;
  if ((threadIdx.x & 31) == 0) { smn[wave] = mn; smx[wave] = mx; }
  __syncthreads();
  if (threadIdx.x == 0) {
#pragma unroll
    for (int w = 1; w < kThreads / 32; ++w) {
      mn = fminf(mn, smn[w]);
      mx = fmaxf(mx, smx[w]);
    }
    const int idx = s * kBlocksPerSample + blk;
    partial[2 * idx + 0] = mn;
    partial[2 * idx + 1] = mx;
  }
}

// ---------------------------------------------------------------------------
// Pass 2: one small block per sample folds the 98 partials and precomputes
// the loop-invariant quantization constants so pass 3 is division-free:
//   params[s] = { m, rng, 254/rng, rng/254 }   (single s_load_b128 in pass 3)
// ---------------------------------------------------------------------------
__global__ __launch_bounds__(128) void qr6_final_minmax(
    const float* __restrict__ partial, v4f* __restrict__ params) {
  const int s = (int)blockIdx.x;
  const float* __restrict__ src = partial + (size_t)s * kBlocksPerSample * 2;
  float mn =  QR6_FLT_MAX;
  float mx = -QR6_FLT_MAX;
  for (int i = threadIdx.x; i < kBlocksPerSample; i += 128) {
    mn = fminf(mn, src[2 * i + 0]);
    mx = fmaxf(mx, src[2 * i + 1]);
  }
#pragma unroll
  for (int off = 16; off > 0; off >>= 1) {
    mn = fminf(mn, __shfl_xor(mn, off, 32));
    mx = fmaxf(mx, __shfl_xor(mx, off, 32));
  }
  __shared__ float smn[4];
  __shared__ float smx[4];
  const int wave = threadIdx.x >> 5;
  if ((threadIdx.x & 31) == 0) { smn[wave] = mn; smx[wave] = mx; }
  __syncthreads();
  if (threadIdx.x == 0) {
#pragma unroll
    for (int w = 1; w < 4; ++w) {
      mn = fminf(mn, smn[w]);
      mx = fmaxf(mx, smx[w]);
    }
    const float rng = mx - mn;
    v4f prm;
    prm.x = mn;                 // m
    prm.y = rng;                // rng
    prm.z = 254.0f / rng;       // qscale  (forward quant scale)
    prm.w = rng / 254.0f;       // dq      (dequant scale)
    params[s] = prm;
  }
}

// Division-free per-element fake-quant + ReLU6. Matches reference operation
// order to ~1 ulp: 254*(x-m)/rng -> (x-m)*qscale; rng/254 -> dq;
// the /1000 then *1000 non-cancellation is preserved explicitly.
__device__ __forceinline__ float qr6_one(float x, float m, float qscale, float dq) {
  float qn = rintf((x - m) * qscale - 127.0f);   // integer-valued in [-127,127]
  float q  = qn * 1.0e-3f;                       // "/ 1000"
  float y  = (1000.0f * q + 127.0f) * dq + m;
  return fminf(fmaxf(y, 0.0f), 6.0f);
}

// ---------------------------------------------------------------------------
// Pass 3: fake-quant + ReLU6 with double-buffered async global->LDS staging
// (gfx1250 ASYNCcnt path). Blocks walk in REVERSE order vs pass 1 so the
// hottest tail of x (still resident in the 192MB L2) is consumed first.
// Output uses non-temporal stores so the streamed result doesn't evict x.
// Each lane stages its own 16B slot, so no barrier is needed between the
// async completion wait and the LDS read.
// ---------------------------------------------------------------------------
__global__ __launch_bounds__(kThreads) void qr6_transform(
    const float* __restrict__ x, const v4f* __restrict__ params,
    float* __restrict__ out) {
  const int bid = (int)gridDim.x - 1 - (int)blockIdx.x;   // reverse for L2 reuse
  const int s   = bid / kBlocksPerSample;
  const int blk = bid % kBlocksPerSample;
  const size_t base = (size_t)s * kV4Per + (size_t)blk * (kThreads * kV4PerThread);
  const v4f* __restrict__ src = reinterpret_cast<const v4f*>(x) + base;
  v4f* __restrict__ dst = reinterpret_cast<v4f*>(out) + base;

  const v4f prm = params[s];         // uniform -> scalar s_load_b128
  const float m      = prm.x;
  const float qscale = prm.z;
  const float dq     = prm.w;

  __shared__ v4f tile[2][kThreads];
  const int tid = threadIdx.x;
  const uint32_t lds0 = (uint32_t)(uintptr_t)(&tile[0][tid]);
  const uint32_t lds1 = (uint32_t)(uintptr_t)(&tile[1][tid]);

  // Prologue: stage chunk 0 into buffer 0.
  {
    const uint64_t g = (uint64_t)(uintptr_t)(src + 0 * kThreads + tid);
    asm volatile("global_load_async_to_lds_b128 %0, %1, off"
                 :: "v"(lds0), "v"(g) : "memory");
  }
#pragma unroll
  for (int i = 0; i < kV4PerThread; ++i) {
    if (i + 1 < kV4PerThread) {
      // Kick off next chunk into the other buffer, then wait for the
      // in-flight count to drop to 1: async loads complete in order, so
      // the current buffer is guaranteed resident in LDS.
      const uint32_t nxt_lds = (i & 1) ? lds0 : lds1;
      const uint64_t g = (uint64_t)(uintptr_t)(src + (i + 1) * kThreads + tid);
      asm volatile("global_load_async_to_lds_b128 %0, %1, off"
                   :: "v"(nxt_lds), "v"(g) : "memory");
      asm volatile("s_wait_asynccnt 0x1" ::: "memory");
    } else {
      asm volatile("s_wait_asynccnt 0x0" ::: "memory");
    }
    v4f v = tile[i & 1][tid];   // ds_load_b128
    v4f r;
    r.x = qr6_one(v.x, m, qscale, dq);
    r.y = qr6_one(v.y, m, qscale, dq);
    r.z = qr6_one(v.z, m, qscale, dq);
    r.w = qr6_one(v.w, m, qscale, dq);
    __builtin_nontemporal_store(r, &dst[i * kThreads + tid]);
  }
}

extern "C" void kernel_launch(void* const* d_in, const int* in_sizes, int n_in,
                              void* d_out, int out_size, void* d_ws, size_t ws_size,
                              hipStream_t stream) {
  const float* x = (const float*)d_in[0];
  float* out = (float*)d_out;

  // Workspace layout: [partial: 64*98*2 f32 = 50176 B][params: 64 x float4]
  float* partial = (float*)d_ws;
  v4f*   params  = reinterpret_cast<v4f*>(
      (char*)d_ws + (size_t)kNumSamples * kBlocksPerSample * 2 * sizeof(float));

  const int grid = kNumSamples * kBlocksPerSample;   // 6272 blocks
  qr6_partial_minmax<<<grid, kThreads, 0, stream>>>(x, partial);
  qr6_final_minmax<<<kNumSamples, 128, 0, stream>>>(partial, params);
  qr6_transform<<<grid, kThreads, 0, stream>>>(x, params, out);
}